// SemanticEdgeGating_18683107737872
// MI455X (gfx1250) — compile-verified
//
#include <hip/hip_runtime.h>

typedef __attribute__((ext_vector_type(2))) float v2f;
typedef __attribute__((ext_vector_type(8))) float v8f;

#define BB   8
#define AAC  192
#define DD   128
#define RELN 5
#define HHC  128
#define ROWS (BB * AAC)                 // 1536
#define NEDGE ((size_t)BB * AAC * AAC)  // 294912

// ---------------------------------------------------------------------------
// Kernel 0: Pg[b][h] = graph_context[b,:] @ W1[773:901, h] + b1[h]
// Tiny: 8 blocks x 128 threads.
// ---------------------------------------------------------------------------
__global__ void pg_kernel(const float* __restrict__ gc,
                          const float* __restrict__ W1,
                          const float* __restrict__ b1,
                          float* __restrict__ Pg) {
    int b = blockIdx.x;
    int h = threadIdx.x;
    float acc = b1[h];
    const float* w = W1 + (size_t)(6 * DD + RELN) * HHC + h;  // row 773
    const float* g = gc + (size_t)b * DD;
#pragma unroll 8
    for (int d = 0; d < DD; ++d) acc = fmaf(g[d], w[(size_t)d * HHC], acc);
    Pg[b * HHC + h] = acc;
}

// ---------------------------------------------------------------------------
// Kernel 1: Pcat[row, 0:128]  = [coord|goal|frontier](row) @ W1(i-rows)
//           Pcat[row, 128:256]= [coord|goal|frontier](row) @ W1(j-rows)
// One wave32 per 16x16 output tile, fp32 WMMA, K = 384 in steps of 4.
// W1 row map: i-part -> kk + blk*128 ; j-part -> kk + blk*128 + 128
// ---------------------------------------------------------------------------
__global__ void pcat_wmma_kernel(const float* __restrict__ coord,
                                 const float* __restrict__ goal,
                                 const float* __restrict__ frontier,
                                 const float* __restrict__ W1,
                                 float* __restrict__ Pcat) {
    const int tile = blockIdx.x;       // 96 * 16 = 1536 tiles
    const int mt   = tile >> 4;        // M tile (rows of Pcat)
    const int nt   = tile & 15;        // N tile (cols of Pcat)
    const int lane = threadIdx.x;      // 0..31, full wave (EXEC all ones)

    const int row   = mt * 16 + (lane & 15);   // A-matrix row for this lane
    const int n     = nt * 16 + (lane & 15);   // B/C column for this lane
    const int khalf = (lane >> 4) * 2;         // A/B K sub-offset: 0 or 2
    const int jsel  = (n >= HHC) ? 1 : 0;
    const int col   = n & (HHC - 1);

    v8f c = {};
    for (int blk = 0; blk < 3; ++blk) {
        const float* x = (blk == 0 ? coord : (blk == 1 ? goal : frontier))
                         + (size_t)row * DD + khalf;
        const float* wp = W1 + (size_t)(blk * DD + jsel * DD + khalf) * HHC + col;
#pragma unroll 4
        for (int k = 0; k < DD; k += 4) {
            v2f a;  a.x  = x[k];
                    a.y  = x[k + 1];
            v2f bf; bf.x = wp[(size_t)k * HHC];
                    bf.y = wp[(size_t)(k + 1) * HHC];
            // D = A(16x4 f32) * B(4x16 f32) + C  -> v_wmma_f32_16x16x4_f32
            c = __builtin_amdgcn_wmma_f32_16x16x4_f32(
                    false, a, false, bf, (short)0, c, false, false);
        }
    }

    // C/D layout: VGPR v -> M = Mbase + v (lanes 0-15) / Mbase + 8 + v (lanes 16-31)
    const int mbase = mt * 16 + ((lane >= 16) ? 8 : 0);
#pragma unroll
    for (int v = 0; v < 8; ++v) {
        Pcat[(size_t)(mbase + v) * 256 + n] = c[v];
    }
}

// ---------------------------------------------------------------------------
// Kernel 2: fused per-edge MLP + masked softmax per (b,i) row.
// One workgroup per row: 192 threads (6 full wave32s), thread j owns edge (b,i,j).
// ---------------------------------------------------------------------------
__global__ void edge_kernel(const float* __restrict__ rel_feat,
                            const int*   __restrict__ rule_adj,
                            const float* __restrict__ base_score,
                            const float* __restrict__ W1,
                            const float* __restrict__ W2,
                            const float* __restrict__ b2,
                            const float* __restrict__ Pcat,
                            const float* __restrict__ Pg,
                            float* __restrict__ out) {
    __shared__ float s_base[HHC];          // Pi[b,i,:] + Pg[b,:] (includes b1)
    __shared__ float s_w1r[RELN * HHC];    // W1 rows 768..772
    __shared__ float s_w2[HHC];
    __shared__ float s_red[8];
    __shared__ float s_bc;

    const int bi  = blockIdx.x;            // b*A + i
    const int b   = bi / AAC;
    const int tid = threadIdx.x;           // 0..191  == j

    if (tid < HHC) {
        s_base[tid] = Pcat[(size_t)bi * 256 + tid] + Pg[b * HHC + tid];
        s_w2[tid]   = W2[tid];
    }
    for (int idx = tid; idx < RELN * HHC; idx += AAC)
        s_w1r[idx] = W1[(size_t)(6 * DD) * HHC + idx];   // rows 768.. contiguous
    __syncthreads();

    const size_t edge = (size_t)bi * AAC + tid;
    const float* rf = rel_feat + edge * RELN;
    const float rf0 = rf[0], rf1 = rf[1], rf2 = rf[2], rf3 = rf[3], rf4 = rf[4];
    const float* pj = Pcat + (size_t)(b * AAC + tid) * 256 + HHC;

    float learned = 0.f;
#pragma unroll 4
    for (int h = 0; h < HHC; ++h) {
        float hv = s_base[h] + pj[h];
        hv = fmaf(rf0, s_w1r[h],           hv);
        hv = fmaf(rf1, s_w1r[HHC + h],     hv);
        hv = fmaf(rf2, s_w1r[2 * HHC + h], hv);
        hv = fmaf(rf3, s_w1r[3 * HHC + h], hv);
        hv = fmaf(rf4, s_w1r[4 * HHC + h], hv);
        hv = fmaxf(hv, 0.f);
        learned = fmaf(hv, s_w2[h], learned);
    }
    learned += b2[0];

    const float sc_raw = base_score[edge] + learned;
    const bool  adj    = rule_adj[edge] != 0;
    const float sc     = adj ? sc_raw : -1e9f;

    out[2 * NEDGE + edge] = learned;   // learned (pre-mask)
    out[NEDGE + edge]     = sc;        // score (masked, as in reference)

    // ---- row max ----
    float m = sc;
    for (int off = 16; off > 0; off >>= 1) m = fmaxf(m, __shfl_xor(m, off, 32));
    if ((tid & 31) == 0) s_red[tid >> 5] = m;
    __syncthreads();
    if (tid == 0) {
        float mm = s_red[0];
        for (int w = 1; w < AAC / 32; ++w) mm = fmaxf(mm, s_red[w]);
        s_bc = mm;
    }
    __syncthreads();

    // ---- softmax ----
    const float e = expf(sc - s_bc);
    float ssum = e;
    for (int off = 16; off > 0; off >>= 1) ssum += __shfl_xor(ssum, off, 32);
    __syncthreads();                      // protect s_red reuse
    if ((tid & 31) == 0) s_red[tid >> 5] = ssum;
    __syncthreads();
    if (tid == 0) {
        float t = 0.f;
        for (int w = 0; w < AAC / 32; ++w) t += s_red[w];
        s_bc = t;
    }
    __syncthreads();
    float wgt = e / s_bc;
    wgt = adj ? wgt : 0.f;

    // ---- renormalize over masked weights ----
    float rs = wgt;
    for (int off = 16; off > 0; off >>= 1) rs += __shfl_xor(rs, off, 32);
    __syncthreads();
    if ((tid & 31) == 0) s_red[tid >> 5] = rs;
    __syncthreads();
    if (tid == 0) {
        float t = 0.f;
        for (int w = 0; w < AAC / 32; ++w) t += s_red[w];
        s_bc = t;
    }
    __syncthreads();

    out[edge] = wgt / (s_bc + 1e-8f);
}

// ---------------------------------------------------------------------------
extern "C" void kernel_launch(void* const* d_in, const int* in_sizes, int n_in,
                              void* d_out, int out_size, void* d_ws, size_t ws_size,
                              hipStream_t stream) {
    (void)in_sizes; (void)n_in; (void)out_size; (void)ws_size;
    const float* coord    = (const float*)d_in[0];
    const float* goal     = (const float*)d_in[1];
    const float* frontier = (const float*)d_in[2];
    const float* rel_feat = (const float*)d_in[3];
    const int*   rule_adj = (const int*)  d_in[4];
    const float* base_sc  = (const float*)d_in[5];
    const float* gc       = (const float*)d_in[6];
    const float* W1       = (const float*)d_in[7];
    const float* b1       = (const float*)d_in[8];
    const float* W2       = (const float*)d_in[9];
    const float* b2       = (const float*)d_in[10];

    float* Pcat = (float*)d_ws;                    // 1536 * 256 f32 = 1.5 MB
    float* Pg   = Pcat + (size_t)ROWS * 256;       // 8 * 128 f32

    pg_kernel<<<BB, HHC, 0, stream>>>(gc, W1, b1, Pg);
    pcat_wmma_kernel<<<ROWS, 32, 0, stream>>>(coord, goal, frontier, W1, Pcat);
    edge_kernel<<<ROWS, AAC, 0, stream>>>(rel_feat, rule_adj, base_sc, W1, W2, b2,
                                          Pcat, Pg, (float*)d_out);
}